// ConvTemporalGraphical_31971736551434
// MI455X (gfx1250) — compile-verified
//
#include <hip/hip_runtime.h>

typedef __attribute__((ext_vector_type(2))) float v2f;
typedef __attribute__((ext_vector_type(8))) float v8f;

#define WMMA_F32(a, b, c) \
  __builtin_amdgcn_wmma_f32_16x16x4_f32(false, (a), false, (b), (short)0, (c), false, false)

constexpr int N_ = 32, CIN_ = 64, COUT_ = 64, T_ = 16, V_ = 256, K_ = 5;
constexpr int PIX_ = T_ * V_;  // 4096 pixels per (n, channel)

// ---------------------------------------------------------------------------
// Build the B-operand (v2f) for k-step s (covering channels 4s..4s+3) from a
// 16ch x 16pix tile stored in WMMA C/D layout:
//   value(ch, pix) lives in lane (pix + 16*(ch>>3)), register (ch & 7).
// Target B layout: lane l (l<16):  {d[4s+0, l], d[4s+1, l]}
//                  lane l (l>=16): {d[4s+2, l&15], d[4s+3, l&15]}
// ---------------------------------------------------------------------------
__device__ __forceinline__ v2f b_from_d(const float* d, int s, int lane) {
  const int p = lane & 15;
  v2f b;
  if (s < 2) {  // channels 0..7 live in low-half lanes
    float tx = __shfl(d[4 * s + 2], p, 32);
    float ty = __shfl(d[4 * s + 3], p, 32);
    b.x = (lane < 16) ? d[4 * s + 0] : tx;
    b.y = (lane < 16) ? d[4 * s + 1] : ty;
  } else {      // channels 8..15 live in high-half lanes, reg (ch-8)
    float tx = __shfl(d[4 * s - 8 + 0], p + 16, 32);
    float ty = __shfl(d[4 * s - 8 + 1], p + 16, 32);
    b.x = (lane < 16) ? tx : d[4 * s - 8 + 2];
    b.y = (lane < 16) ? ty : d[4 * s - 8 + 3];
  }
  return b;
}

// ---------------------------------------------------------------------------
// Kernel 1: xc[n,o,t,v] = sum_c conv_w[o,c] * x[n,c,t,v] + conv_b[o]
// One wave = one 16-pixel tile, 4 M-tiles (64 out ch), K=64 in 16 steps.
// ---------------------------------------------------------------------------
__global__ __launch_bounds__(256) void k_pointwise_conv(
    const float* __restrict__ x, const float* __restrict__ w,
    const float* __restrict__ b, float* __restrict__ xc) {
  const int lane = threadIdx.x & 31;
  const int wave = threadIdx.x >> 5;
  const int n = blockIdx.x;
  const int tile = blockIdx.y * 8 + wave;  // 0..255
  const int pix0 = tile * 16;
  const int hh = lane >> 4;
  const int row = lane & 15;
  const float* xn = x + (size_t)n * CIN_ * PIX_;

  v8f acc[4] = {};
#pragma unroll
  for (int s = 0; s < 16; ++s) {
    const int c0 = 4 * s + 2 * hh;
    v2f bop;
    bop.x = xn[(size_t)(c0 + 0) * PIX_ + pix0 + row];
    bop.y = xn[(size_t)(c0 + 1) * PIX_ + pix0 + row];
#pragma unroll
    for (int m = 0; m < 4; ++m) {
      v2f aop = *(const v2f*)(w + (m * 16 + row) * CIN_ + c0);  // 8B aligned
      acc[m] = WMMA_F32(aop, bop, acc[m]);
    }
  }
#pragma unroll
  for (int m = 0; m < 4; ++m) {
#pragma unroll
    for (int r = 0; r < 8; ++r) {
      const int o = m * 16 + r + 8 * hh;
      xc[((size_t)n * COUT_ + o) * PIX_ + pix0 + row] = acc[m][r] + b[o];
    }
  }
}

// ---------------------------------------------------------------------------
// Kernel 2: Dl[n,w] = 1 / (sum_{k,v} A[n,k,v,w] + 0.001)
// ---------------------------------------------------------------------------
__global__ void k_degree(const float* __restrict__ A, float* __restrict__ Dl) {
  const int idx = blockIdx.x * blockDim.x + threadIdx.x;  // n*256 + w
  if (idx >= N_ * V_) return;
  const int n = idx >> 8, w = idx & 255;
  const float* An = A + (size_t)n * K_ * V_ * V_ + w;
  float s = 0.f;
  for (int kv = 0; kv < K_ * V_; ++kv) s += An[(size_t)kv * V_];
  Dl[idx] = 1.0f / (s + 0.001f);
}

// ---------------------------------------------------------------------------
// Kernel 3 (fused): per workgroup = (n, 16-wide w tile).
//   For each 16-wide v chunk:
//     producer: each wave runs the (mask-selected) MLP for 2 w-columns x 16 v
//               -> A_eff tile (64ch x 16v x 16w) scaled by Dl*mask -> LDS
//     consumer: each wave accumulates 8 channels of
//               out[c, t(16), w(16)] += xc[c, t, v(16)] @ A_eff[c, v, w]
// ---------------------------------------------------------------------------
__global__ __launch_bounds__(256) void k_fused(
    const float* __restrict__ A, const float* __restrict__ xc,
    const float* __restrict__ Dl, const float* __restrict__ hmask,
    const float* __restrict__ nhmask,
    const float* __restrict__ nw1, const float* __restrict__ nb1,
    const float* __restrict__ nw2, const float* __restrict__ nb2,
    const float* __restrict__ nw3, const float* __restrict__ nb3,
    const float* __restrict__ hw1, const float* __restrict__ hb1,
    const float* __restrict__ hw2, const float* __restrict__ hb2,
    const float* __restrict__ hw3, const float* __restrict__ hb3,
    float* __restrict__ out) {
  __shared__ float Alds[64][16][16];  // [c][w_local][v_local] = 64 KB

  const int lane = threadIdx.x & 31;
  const int wave = threadIdx.x >> 5;
  const int n = blockIdx.x;
  const int wbase = blockIdx.y * 16;
  const int hh = lane >> 4;
  const int row = lane & 15;
  const float* An = A + (size_t)n * K_ * V_ * V_;

  v8f acc[8] = {};  // out accumulators: 8 channels x (16t x 16w)

  for (int vb = 0; vb < V_; vb += 16) {
    __syncthreads();  // previous einsum readers done before overwrite

    // ---------------- producer: MLP for 2 w-columns per wave ---------------
#pragma unroll
    for (int tt = 0; tt < 2; ++tt) {
      const int wl = wave * 2 + tt;  // 0..15
      const int wg = wbase + wl;
      const float hm = hmask[n * V_ + wg];
      const bool human = hm > 0.5f;
      const float mval = human ? hm : nhmask[n * V_ + wg];
      const float* W1 = human ? hw1 : nw1; const float* B1 = human ? hb1 : nb1;
      const float* W2 = human ? hw2 : nw2; const float* B2 = human ? hb2 : nb2;
      const float* W3 = human ? hw3 : nw3; const float* B3 = human ? hb3 : nb3;
      const float dl = Dl[n * V_ + wg] * mval;
      const int v = vb + row;  // pixel (v, wg); 16 pixels across lanes

      // ---- layer 1: 16ch = W1(16x5) @ A(5 x 16pix), K padded 5->8
      float h1a[8];
      {
        const int c0 = 2 * hh;
        v2f a0, a1, b0, b1;
        b0.x = An[(size_t)(c0 + 0) * (V_ * V_) + (size_t)v * V_ + wg];
        b0.y = An[(size_t)(c0 + 1) * (V_ * V_) + (size_t)v * V_ + wg];
        a0.x = W1[row * 5 + c0];
        a0.y = W1[row * 5 + c0 + 1];
        const float a4 = An[(size_t)4 * (V_ * V_) + (size_t)v * V_ + wg];
        b1.x = hh ? 0.f : a4;   b1.y = 0.f;          // K=4 real, K=5..7 zero
        a1.x = hh ? 0.f : W1[row * 5 + 4];  a1.y = 0.f;
        v8f c = {};
        c = WMMA_F32(a0, b0, c);
        c = WMMA_F32(a1, b1, c);
#pragma unroll
        for (int r = 0; r < 8; ++r)
          h1a[r] = fmaxf(c[r] + B1[r + 8 * hh], 0.f);
      }

      // ---- layer 2: 32ch = W2(32x16) @ h1(16 x 16pix)
      float h2a[2][8];
#pragma unroll
      for (int m = 0; m < 2; ++m) {
        v8f c = {};
#pragma unroll
        for (int s = 0; s < 4; ++s) {
          v2f aop = *(const v2f*)(W2 + (m * 16 + row) * 16 + 4 * s + 2 * hh);
          v2f bop = b_from_d(h1a, s, lane);
          c = WMMA_F32(aop, bop, c);
        }
#pragma unroll
        for (int r = 0; r < 8; ++r)
          h2a[m][r] = fmaxf(c[r] + B2[m * 16 + r + 8 * hh], 0.f);
      }

      // ---- layer 3: 64ch = W3(64x32) @ h2(32 x 16pix); relu, *Dl*mask -> LDS
#pragma unroll
      for (int m = 0; m < 4; ++m) {
        v8f c = {};
#pragma unroll
        for (int s = 0; s < 8; ++s) {
          v2f aop = *(const v2f*)(W3 + (m * 16 + row) * 32 + 4 * s + 2 * hh);
          v2f bop = b_from_d((s < 4) ? h2a[0] : h2a[1], s & 3, lane);
          c = WMMA_F32(aop, bop, c);
        }
#pragma unroll
        for (int r = 0; r < 8; ++r) {
          const int ch = m * 16 + r + 8 * hh;
          Alds[ch][wl][row] = fmaxf(c[r] + B3[ch], 0.f) * dl;
        }
      }
    }
    __syncthreads();

    // ---------------- consumer: out[c,t,w] += xc[c,t,v] @ Aeff[c,v,w] ------
#pragma unroll
    for (int cl = 0; cl < 8; ++cl) {
      const int c = wave * 8 + cl;
      const float* xrow = xc + ((size_t)n * COUT_ + c) * PIX_;
#pragma unroll
      for (int s = 0; s < 4; ++s) {
        const int kv = 4 * s + 2 * hh;
        v2f aop = *(const v2f*)(xrow + (size_t)row * V_ + vb + kv);  // t=row
        v2f bop = *(const v2f*)(&Alds[c][row][kv]);                  // w=row
        acc[cl] = WMMA_F32(aop, bop, acc[cl]);
      }
    }
  }

  // store out[n,c,t,wbase+w]
#pragma unroll
  for (int cl = 0; cl < 8; ++cl) {
    const int c = wave * 8 + cl;
#pragma unroll
    for (int r = 0; r < 8; ++r) {
      const int t = r + 8 * hh;
      out[(((size_t)n * COUT_ + c) * T_ + t) * V_ + wbase + row] = acc[cl][r];
    }
  }
}

// ---------------------------------------------------------------------------
extern "C" void kernel_launch(void* const* d_in, const int* in_sizes, int n_in,
                              void* d_out, int out_size, void* d_ws,
                              size_t ws_size, hipStream_t stream) {
  (void)in_sizes; (void)n_in; (void)out_size; (void)ws_size;
  const float* x   = (const float*)d_in[0];
  const float* A   = (const float*)d_in[1];
  const float* hm  = (const float*)d_in[2];
  const float* nhm = (const float*)d_in[3];
  const float* cw  = (const float*)d_in[4];
  const float* cb  = (const float*)d_in[5];
  const float* nw1 = (const float*)d_in[6];  const float* nb1 = (const float*)d_in[7];
  const float* nw2 = (const float*)d_in[8];  const float* nb2 = (const float*)d_in[9];
  const float* nw3 = (const float*)d_in[10]; const float* nb3 = (const float*)d_in[11];
  const float* hw1 = (const float*)d_in[12]; const float* hb1 = (const float*)d_in[13];
  const float* hw2 = (const float*)d_in[14]; const float* hb2 = (const float*)d_in[15];
  const float* hw3 = (const float*)d_in[16]; const float* hb3 = (const float*)d_in[17];

  const size_t out_elems = (size_t)N_ * COUT_ * T_ * V_;   // 8388608
  const size_t a_elems   = (size_t)N_ * K_ * V_ * V_;      // 10485760

  float* xc = (float*)d_ws;            // workspace: xc then Dl (~33.6 MB)
  float* Dl = xc + out_elems;
  float* out = (float*)d_out;
  float* Aout = out + out_elems;

  // Second tuple element: pass-through copy of A.
  hipMemcpyAsync(Aout, A, a_elems * sizeof(float), hipMemcpyDeviceToDevice,
                 stream);

  k_pointwise_conv<<<dim3(N_, 32), 256, 0, stream>>>(x, cw, cb, xc);
  k_degree<<<dim3((N_ * V_) / 256), 256, 0, stream>>>(A, Dl);
  k_fused<<<dim3(N_, V_ / 16), 256, 0, stream>>>(
      A, xc, Dl, hm, nhm, nw1, nb1, nw2, nb2, nw3, nb3, hw1, hb1, hw2, hb2,
      hw3, hb3, out);
}